// SpectralBias_12180527251617
// MI455X (gfx1250) — compile-verified
//
#include <hip/hip_runtime.h>
#include <math.h>

typedef __attribute__((ext_vector_type(2))) float v2f;
typedef __attribute__((ext_vector_type(8))) float v8f;

#define LSEQ 2048
#define NH   16
#define DHD  64
#define HID  128

// omegas: logspace(log10(2pi/1e6), log10(2pi/2048), 6) -- computed in double like numpy
__device__ __forceinline__ float omega_k(int k) {
    const double lmin = -11.977623973526158;   // ln(2*pi/1e6)
    const double lmax = -5.786741919750054;    // ln(2*pi/2048)
    double lw = lmin + (double)k * ((lmax - lmin) * 0.2);
    return (float)exp(lw);
}

__device__ __forceinline__ float sigmoidf_(float x) { return 1.0f / (1.0f + expf(-x)); }
__device__ __forceinline__ float softplusf_(float x) {
    // stable: max(x,0) + log1p(exp(-|x|))
    return fmaxf(x, 0.0f) + log1pf(expf(-fabsf(x)));
}

// ---------------- kernel 0: cos/sin means over d in [0, L) ----------------
__global__ void __launch_bounds__(256) sb_means_kernel(float* __restrict__ means) {
    __shared__ float red[256];
    const int tid = threadIdx.x;
    for (int k = 0; k < 6; ++k) {
        const float w = omega_k(k);
        float sc = 0.0f, ss = 0.0f;
        for (int d = tid; d < LSEQ; d += 256) {
            float s, c;
            sincosf(w * (float)d, &s, &c);
            sc += c; ss += s;
        }
        red[tid] = sc; __syncthreads();
        for (int o = 128; o > 0; o >>= 1) {
            if (tid < o) red[tid] += red[tid + o];
            __syncthreads();
        }
        if (tid == 0) means[k] = red[0] * (1.0f / (float)LSEQ);
        __syncthreads();
        red[tid] = ss; __syncthreads();
        for (int o = 128; o > 0; o >>= 1) {
            if (tid < o) red[tid] += red[tid + o];
            __syncthreads();
        }
        if (tid == 0) means[6 + k] = red[0] * (1.0f / (float)LSEQ);
        __syncthreads();
    }
}

// ---------------- kernel 1: B table [16 x L] ----------------
// rows 0..5: cos(w_k * j); rows 6..11: sin(w_k * j); row 12: 1; row 13: j; rows 14,15: 0
__global__ void __launch_bounds__(256) sb_btab_kernel(float* __restrict__ Bt) {
    const int j = blockIdx.x * 256 + threadIdx.x;
    if (j >= LSEQ) return;
#pragma unroll
    for (int k = 0; k < 6; ++k) {
        float s, c;
        sincosf(omega_k(k) * (float)j, &s, &c);
        Bt[k * LSEQ + j]       = c;
        Bt[(6 + k) * LSEQ + j] = s;
    }
    Bt[12 * LSEQ + j] = 1.0f;
    Bt[13 * LSEQ + j] = (float)j;
    Bt[14 * LSEQ + j] = 0.0f;
    Bt[15 * LSEQ + j] = 0.0f;
}

// ---------------- kernel 2: per-query MLP -> 16 A-coefficients ----------------
__global__ void __launch_bounds__(256) sb_coef_kernel(
    const float* __restrict__ q, const float* __restrict__ w1, const float* __restrict__ b1,
    const float* __restrict__ w2, const float* __restrict__ b2,
    const float* __restrict__ means, float* __restrict__ Aout) {

    __shared__ float s_w1[HID * DHD];   // 32 KB
    __shared__ float s_w2[9 * HID];
    __shared__ float s_b1[HID];
    __shared__ float s_b2[12];
    __shared__ float s_mean[12];

    const int tid = threadIdx.x;
    for (int idx = tid; idx < HID * DHD; idx += 256) s_w1[idx] = w1[idx];
    for (int idx = tid; idx < 9 * HID;  idx += 256) s_w2[idx] = w2[idx];
    if (tid < HID) s_b1[tid] = b1[tid];
    if (tid < 9)   s_b2[tid] = b2[tid];
    if (tid < 12)  s_mean[tid] = means[tid];
    __syncthreads();

    const int g = blockIdx.x * 256 + tid;      // 0 .. H*L-1
    const int i = g & (LSEQ - 1);              // position in sequence

    // load this query's 64 features into registers
    float qreg[DHD];
    const float* qp = q + (size_t)g * DHD;
#pragma unroll
    for (int d = 0; d < DHD; ++d) qreg[d] = qp[d];

    // fused MLP: hidden never stored, each silu(h_j) folded into the 9 outputs
    float y[9];
#pragma unroll
    for (int o = 0; o < 9; ++o) y[o] = s_b2[o];
    for (int j = 0; j < HID; ++j) {
        float acc = s_b1[j];
        const float* wr = &s_w1[j * DHD];
#pragma unroll
        for (int d = 0; d < DHD; ++d) acc = fmaf(qreg[d], wr[d], acc);
        float hs = acc * sigmoidf_(acc);   // SiLU
#pragma unroll
        for (int o = 0; o < 9; ++o) y[o] = fmaf(hs, s_w2[o * HID + j], y[o]);
    }

    float coefC[6] = {0,0,0,0,0,0};
    float coefS[6] = {0,0,0,0,0,0};
#pragma unroll
    for (int m = 0; m < 2; ++m) {
        float p  = sigmoidf_(y[m]);
        float ds = sigmoidf_(y[2 + m]) * 2047.0f;                // delta*
        float sg = 32.0f + sigmoidf_(y[4 + m]) * 224.0f;         // sigma
        float cc = 0.01f * softplusf_(y[6 + m]);                 // amplitude
        float a[6], asum = 0.0f;
#pragma unroll
        for (int k = 0; k < 6; ++k) {
            float w = omega_k(k);
            a[k] = expf(-0.5f * w * w * sg * sg);
            asum += a[k];
        }
        float scale = (p * cc) / (asum + 1e-9f);
#pragma unroll
        for (int k = 0; k < 6; ++k) {
            float sph, cph;
            sincosf(omega_k(k) * ds, &sph, &cph);
            float amp = a[k] * scale;
            coefC[k] = fmaf(amp, cph, coefC[k]);
            coefS[k] = fmaf(amp, sph, coefS[k]);
        }
    }
    float slope = softplusf_(y[8]);
    float b0 = 0.0f;
#pragma unroll
    for (int k = 0; k < 6; ++k) b0 += coefC[k] * s_mean[k] + coefS[k] * s_mean[6 + k];
    float sOverTau = slope * (1.0f / 64.0f);

    float* Ar = Aout + (size_t)g * 16;
#pragma unroll
    for (int k = 0; k < 6; ++k) {
        float si, ci;
        sincosf(omega_k(k) * (float)i, &si, &ci);
        Ar[k]     = coefC[k] * ci + coefS[k] * si;   // U
        Ar[6 + k] = coefC[k] * si - coefS[k] * ci;   // V
    }
    Ar[12] = -(b0 + sOverTau * (float)i);
    Ar[13] = sOverTau;
    Ar[14] = 0.0f;
    Ar[15] = 0.0f;
}

// ---------------- kernel 3: rank-16 GEMM via V_WMMA_F32_16X16X4_F32 ----------------
// grid: (128 i-tiles, 16 heads), block: 256 threads = 8 waves.
// Each wave computes 16 j-tiles of 16x16 each (its A operands stay in registers).
__global__ void __launch_bounds__(256) sb_bias_gemm(
    const float* __restrict__ A, const float* __restrict__ Bt, float* __restrict__ out) {

    const int itile = blockIdx.x;        // 0..127
    const int h     = blockIdx.y;        // 0..15
    const int wave  = threadIdx.x >> 5;  // 0..7
    const int lane  = threadIdx.x & 31;
    const int m     = lane & 15;         // row-within-tile for A / col for B,D
    const int hi    = lane >> 4;         // K-pair / row-half selector
    const int i0    = itile * 16;

    const float* Ah = A + ((size_t)h * LSEQ + i0) * 16;

    // A operands: 4 chunks of 16x4.  lane<16 -> K{0,1}, lane>=16 -> K{2,3}
    v2f a[4];
#pragma unroll
    for (int kk = 0; kk < 4; ++kk) {
        const float* ap = Ah + (size_t)m * 16 + kk * 4 + hi * 2;
        v2f av; av.x = ap[0]; av.y = ap[1];
        a[kk] = av;
    }

    float* outh = out + (size_t)h * LSEQ * LSEQ + (size_t)i0 * LSEQ;

    for (int t = 0; t < 16; ++t) {
        const int jtile = wave + t * 8;          // 0..127
        const int j0 = jtile * 16;
        float* otile = outh + j0;

        if (j0 > i0 + 15) {
            // strictly upper-triangular tile: all zeros (d_out is poisoned -> must store)
#pragma unroll
            for (int vr = 0; vr < 8; ++vr)
                otile[(size_t)(vr + 8 * hi) * LSEQ + m] = 0.0f;
            continue;
        }

        v8f c = {0.f,0.f,0.f,0.f,0.f,0.f,0.f,0.f};
#pragma unroll
        for (int kk = 0; kk < 4; ++kk) {
            // B chunk 4x16: lane<16 -> rows {4kk,4kk+1}, lane>=16 -> rows {4kk+2,4kk+3}
            const float* bp = Bt + (size_t)(kk * 4 + hi * 2) * LSEQ + j0 + m;
            v2f bv; bv.x = bp[0]; bv.y = bp[LSEQ];
            c = __builtin_amdgcn_wmma_f32_16x16x4_f32(
                    false, a[kk], false, bv, (short)0, c, false, false);
        }

        const int col = j0 + m;
#pragma unroll
        for (int vr = 0; vr < 8; ++vr) {
            const int row = i0 + vr + 8 * hi;
            float v = (row >= col) ? c[vr] : 0.0f;
            otile[(size_t)(vr + 8 * hi) * LSEQ + m] = v;
        }
    }
}

extern "C" void kernel_launch(void* const* d_in, const int* in_sizes, int n_in,
                              void* d_out, int out_size, void* d_ws, size_t ws_size,
                              hipStream_t stream) {
    const float* q  = (const float*)d_in[0];
    const float* w1 = (const float*)d_in[1];
    const float* b1 = (const float*)d_in[2];
    const float* w2 = (const float*)d_in[3];
    const float* b2 = (const float*)d_in[4];

    float* ws    = (float*)d_ws;
    float* means = ws;                       // 16 floats (12 used)
    float* Btab  = ws + 16;                  // 16*2048 = 32768 floats
    float* Amat  = ws + 16 + 16 * LSEQ;      // 32768*16 = 524288 floats (~2.1 MB total)

    sb_means_kernel<<<1, 256, 0, stream>>>(means);
    sb_btab_kernel<<<LSEQ / 256, 256, 0, stream>>>(Btab);
    sb_coef_kernel<<<(NH * LSEQ) / 256, 256, 0, stream>>>(q, w1, b1, w2, b2, means, Amat);
    sb_bias_gemm<<<dim3(LSEQ / 16, NH), 256, 0, stream>>>(Amat, Btab, (float*)d_out);
}